// HighDimAELoss_37632503447760
// MI455X (gfx1250) — compile-verified
//
#include <hip/hip_runtime.h>
#include <math.h>

// Problem constants (from reference)
#define B_   16
#define MP   30          // people
#define NJ   17          // joints
#define AD   17          // tag dims
#define RES_ 512
#define RR   (RES_ * RES_)
#define MJ   (MP * NJ)        // 510
#define MJD  (MP * NJ * AD)   // 8670

#define AS1 __attribute__((address_space(1)))
#define AS3 __attribute__((address_space(3)))

// CDNA5 async global->LDS gather path (ASYNCcnt-tracked). Guarded so host
// parse and older toolchains fall back to a plain load; device pass on
// gfx1250 takes the async path (confirmed present by round-1 diagnostics).
#if defined(__AMDGCN__) && __has_builtin(__builtin_amdgcn_global_load_async_to_lds_b32)
#define USE_ASYNC_LDS 1
#else
#define USE_ASYNC_LDS 0
#endif

__device__ __forceinline__ void wait_async_zero() {
#if defined(__AMDGCN__)
#if __has_builtin(__builtin_amdgcn_s_wait_asynccnt)
    __builtin_amdgcn_s_wait_asynccnt(0);
#else
    asm volatile("s_wait_asynccnt 0" ::: "memory");
#endif
#endif
}

__global__ __launch_bounds__(256) void ae_loss_kernel(
    const float* __restrict__ tags,    // [B, AD, RES, RES]
    const int*   __restrict__ joints,  // [B, MP, NJ, 2]
    float*       __restrict__ out)     // [2*B] : push[B] then pull[B]
{
    __shared__ float t_s[MJD];     // gathered tags [m][j][d]
    __shared__ float valid_s[MJ];  // per (m,j) visibility
    __shared__ int   xy_s[MJ];     // x*RES + y spatial offset
    __shared__ float sq_s[MJ];     // per (m,j) pairwise-dim sq sum (unweighted)
    __shared__ float mean_s[MP * AD];
    __shared__ float pullp_s[MP];
    __shared__ float pv_s[MP];     // person_valid
    __shared__ float acc_push;

    const int b   = blockIdx.x;
    const int tid = threadIdx.x;
    const int nthreads = blockDim.x;

    // ---- Phase 1: decode joints ------------------------------------------
    const int* jb = joints + (size_t)b * MJ * 2;
    for (int e = tid; e < MJ; e += nthreads) {
        int idx = jb[2 * e + 0];
        int vis = jb[2 * e + 1];
        unsigned off = (unsigned)idx % (unsigned)RR;   // idx >= 0 by construction
        int x = (int)(off % (unsigned)RES_);           // first spatial axis
        int y = (int)(off / (unsigned)RES_);
        xy_s[e]    = x * RES_ + y;
        valid_s[e] = (vis > 0) ? 1.0f : 0.0f;
    }
    if (tid == 0) acc_push = 0.0f;
    __syncthreads();

    // ---- Phase 2: scattered gather tags -> LDS (async on CDNA5) ----------
    const float* tags_b = tags + (size_t)b * AD * RR;
    for (int e = tid; e < MJD; e += nthreads) {
        int mj = e / AD;
        int d  = e - mj * AD;
        const float* src = tags_b + (size_t)d * RR + (size_t)xy_s[mj];
#if USE_ASYNC_LDS
        __builtin_amdgcn_global_load_async_to_lds_b32(
            (AS1 int*)(float*)src,
            (AS3 int*)&t_s[e],
            0, 0);
#else
        t_s[e] = *src;
#endif
    }
#if USE_ASYNC_LDS
    wait_async_zero();
#endif
    __syncthreads();

    // ---- Phase 3: per-(m,j) pairwise-dim squared sum ----------------------
    // sum_{d1,d2}(t_d1 - t_d2)^2 = 2*(D*sum(t^2) - (sum t)^2)
    for (int e = tid; e < MJ; e += nthreads) {
        const float* tp = &t_s[e * AD];
        float Q = 0.0f, P = 0.0f;
        for (int d = 0; d < AD; ++d) { float v = tp[d]; Q += v; P += v * v; }
        sq_s[e] = 2.0f * ((float)AD * P - Q * Q);
    }
    __syncthreads();

    // ---- Phase 4: per-person count / mean / pull_p ------------------------
    for (int m = tid; m < MP; m += nthreads) {
        float cnt = 0.0f;
        for (int j = 0; j < NJ; ++j) cnt += valid_s[m * NJ + j];
        float pv   = (cnt > 0.0f) ? 1.0f : 0.0f;
        float safe = fmaxf(cnt, 1.0f);
        float inv  = 1.0f / safe;
        for (int d = 0; d < AD; ++d) {
            float s = 0.0f;
            for (int j = 0; j < NJ; ++j)
                s += valid_s[m * NJ + j] * t_s[(m * NJ + j) * AD + d];
            mean_s[m * AD + d] = s * inv;
        }
        float ps = 0.0f;
        for (int j = 0; j < NJ; ++j) ps += sq_s[m * NJ + j] * valid_s[m * NJ + j];
        pullp_s[m] = ps / ((float)(AD * AD) * safe);
        pv_s[m]    = pv;
    }
    __syncthreads();

    // ---- Phase 5: push term over M*M pairs (diagonal included) -----------
    float local = 0.0f;
    for (int e = tid; e < MP * MP; e += nthreads) {
        int m1 = e / MP, m2 = e - m1 * MP;
        float w = pv_s[m1] * pv_s[m2];
        if (w != 0.0f) {
            float s = 0.0f;
            for (int d = 0; d < AD; ++d) {
                float pd = mean_s[m1 * AD + d] - mean_s[m2 * AD + d];
                s += expf(-pd * pd);
            }
            local += s * (1.0f / (float)AD);
        }
    }
    if (local != 0.0f) atomicAdd(&acc_push, local);
    __syncthreads();

    // ---- Final: reduce + write -------------------------------------------
    if (tid == 0) {
        float n = 0.0f;
        for (int m = 0; m < MP; ++m) n += pv_s[m];
        float nm = fmaxf(n, 1.0f);
        float pull = 0.0f;
        for (int m = 0; m < MP; ++m) pull += pullp_s[m] * pv_s[m];
        pull /= nm;
        float push = (n >= 2.0f) ? (acc_push / (nm * nm)) : 0.0f;
        out[b]      = push;   // push[B] first
        out[B_ + b] = pull;   // then pull[B]
    }
}

extern "C" void kernel_launch(void* const* d_in, const int* in_sizes, int n_in,
                              void* d_out, int out_size, void* d_ws, size_t ws_size,
                              hipStream_t stream) {
    const float* tags   = (const float*)d_in[0];  // [16,17,512,512] f32
    const int*   joints = (const int*)d_in[1];    // [16,30,17,2] i32
    float*       out    = (float*)d_out;          // [32] f32: push[16] ++ pull[16]
    (void)in_sizes; (void)n_in; (void)out_size; (void)d_ws; (void)ws_size;

    ae_loss_kernel<<<B_, 256, 0, stream>>>(tags, joints, out);
}